// DiscriminativeLoss_38981123178509
// MI455X (gfx1250) — compile-verified
//
#include <hip/hip_runtime.h>
#include <hip/hip_bf16.h>

typedef __attribute__((ext_vector_type(2))) float v2f;
typedef __attribute__((ext_vector_type(8))) float v8f;

#define NCLS 19
#define C_CH 32
#define IN_H 128
#define IN_W 128
#define OUT_H 512
#define OUT_W 512
#define NIMG 4

// Workspace layout (floats):
//   [0, 2432)     cls_sum[n][19][32]
//   [2432, 2508)  count[n][19]
//   [2508, 2584)  s2[n][19]
#define WS_SUM(n) ((n)*NCLS*C_CH)
#define WS_CNT    (NIMG*NCLS*C_CH)
#define WS_S2     (WS_CNT + NIMG*NCLS)
#define WS_TOTAL  (WS_S2 + NIMG*NCLS)

__global__ void zero_ws_kernel(float* __restrict__ ws) {
  int i = blockIdx.x * blockDim.x + threadIdx.x;
  if (i < WS_TOTAL) ws[i] = 0.0f;
}

// One thread = one output pixel. On-the-fly clamped bilinear (== jax.image.resize
// half-pixel + edge renorm), per-class stats into LDS, then one HW fp atomic per
// partial into the global workspace.
__global__ __launch_bounds__(256) void accum_kernel(const float* __restrict__ emb,
                                                    const int* __restrict__ lab,
                                                    float* __restrict__ ws) {
  __shared__ float lsum[NCLS * C_CH];
  __shared__ float lcnt[NCLS];
  __shared__ float ls2[NCLS];
  const int t = threadIdx.x;
  for (int i = t; i < NCLS * C_CH; i += 256) lsum[i] = 0.0f;
  if (t < NCLS) { lcnt[t] = 0.0f; ls2[t] = 0.0f; }
  __syncthreads();

  const int blocksPerImg = (OUT_H * OUT_W) / 256;   // 1024
  const int n   = blockIdx.x / blocksPerImg;
  const int pix = (blockIdx.x % blocksPerImg) * 256 + t;
  const int y = pix >> 9;
  const int x = pix & (OUT_W - 1);

  // half-pixel bilinear, scale factor 1/4; clamped taps == jax edge renorm
  float sx = (x + 0.5f) * 0.25f - 0.5f;
  float sy = (y + 0.5f) * 0.25f - 0.5f;
  int ix0 = (int)floorf(sx); float fx = sx - (float)ix0;
  int iy0 = (int)floorf(sy); float fy = sy - (float)iy0;
  int x0 = ix0 < 0 ? 0 : ix0;
  int x1 = (ix0 + 1 > IN_W - 1) ? (IN_W - 1) : ix0 + 1;
  int y0 = iy0 < 0 ? 0 : iy0;
  int y1 = (iy0 + 1 > IN_H - 1) ? (IN_H - 1) : iy0 + 1;
  float w00 = (1.f - fy) * (1.f - fx), w01 = (1.f - fy) * fx;
  float w10 = fy * (1.f - fx),         w11 = fy * fx;

  const int k = lab[(size_t)n * (OUT_H * OUT_W) + pix];   // 0..18
  const float* base = emb + (size_t)n * C_CH * IN_H * IN_W;
  const int o00 = y0 * IN_W + x0, o01 = y0 * IN_W + x1;
  const int o10 = y1 * IN_W + x0, o11 = y1 * IN_W + x1;

  float* rowsum = &lsum[k * C_CH];
  float ss = 0.0f;
  #pragma unroll 8
  for (int c = 0; c < C_CH; ++c) {
    const float* p = base + c * (IN_H * IN_W);
    float e = w00 * p[o00] + w01 * p[o01] + w10 * p[o10] + w11 * p[o11];
    ss = fmaf(e, e, ss);
    atomicAdd(&rowsum[c], e);              // ds_add_f32
  }
  atomicAdd(&ls2[k], ss);
  atomicAdd(&lcnt[k], 1.0f);
  __syncthreads();

  for (int i = t; i < NCLS * C_CH; i += 256)
    unsafeAtomicAdd(&ws[WS_SUM(n) + i], lsum[i]);          // global_atomic_add_f32
  if (t < NCLS) {
    unsafeAtomicAdd(&ws[WS_CNT + n * NCLS + t], lcnt[t]);
    unsafeAtomicAdd(&ws[WS_S2  + n * NCLS + t], ls2[t]);
  }
}

// One wave per image. Gram matrix G = M * M^T via V_WMMA_F32_16X16X4_F32
// (fp32 WMMA -> exact f32 math), then tiny scalar epilogue.
__global__ __launch_bounds__(32) void finalize_kernel(const float* __restrict__ ws,
                                                      float* __restrict__ out) {
  __shared__ float Mlds[32][33];
  __shared__ float Glds[32][33];
  __shared__ float cnt_s[NCLS];
  __shared__ float s2_s[NCLS];
  const int lane = threadIdx.x;
  const int n = blockIdx.x;
  const float* s   = ws + WS_SUM(n);
  const float* cnt = ws + WS_CNT + n * NCLS;
  const float* s2  = ws + WS_S2  + n * NCLS;
  if (lane < NCLS) { cnt_s[lane] = cnt[lane]; s2_s[lane] = s2[lane]; }
  for (int r = 0; r < 32; ++r) {
    float v = 0.0f;
    if (r < NCLS) v = s[r * 32 + lane] / (cnt[r] + 1.0f);  // class mean, /(count+1)
    Mlds[r][lane] = v;                                     // rows 19..31 zero pad
  }
  __syncthreads();

  const int half = lane >> 4;   // 0: lanes 0-15, 1: lanes 16-31
  const int l16  = lane & 15;
  #pragma unroll
  for (int mt = 0; mt < 2; ++mt) {
    #pragma unroll
    for (int nt = 0; nt < 2; ++nt) {
      v8f acc = {0.f, 0.f, 0.f, 0.f, 0.f, 0.f, 0.f, 0.f};
      #pragma unroll
      for (int kb = 0; kb < 32; kb += 4) {
        // A 16x4: lane half selects K={0,1} vs K={2,3}; row = M = l16
        // B 4x16: same striping, row K, col N = l16; B = M^T so read M rows
        v2f a, b;
        a.x = Mlds[mt * 16 + l16][kb + half * 2 + 0];
        a.y = Mlds[mt * 16 + l16][kb + half * 2 + 1];
        b.x = Mlds[nt * 16 + l16][kb + half * 2 + 0];
        b.y = Mlds[nt * 16 + l16][kb + half * 2 + 1];
        acc = __builtin_amdgcn_wmma_f32_16x16x4_f32(false, a, false, b,
                                                    (short)0, acc, false, false);
      }
      #pragma unroll
      for (int v = 0; v < 8; ++v)   // C/D: VGPR v -> M = v (+8 for high lanes)
        Glds[mt * 16 + v + half * 8][nt * 16 + l16] = acc[v];
    }
  }
  __syncthreads();

  if (lane == 0) {
    float nfg = 0.0f;
    float mask[NCLS];
    for (int k = 1; k < NCLS; ++k) { mask[k] = cnt_s[k] > 0.0f ? 1.0f : 0.0f; nfg += mask[k]; }
    float intra = 0.0f;
    for (int k = 1; k < NCLS; ++k) {
      float c = cnt_s[k];
      // Sum_{p in k} ||e_p - m_k||^2 = S2_k - (c+2)*||m_k||^2 ; ||m_k||^2 = G[k][k]
      float val = (s2_s[k] - (c + 2.0f) * Glds[k][k]) / (32.0f * (c + 1.0f));
      intra += val * mask[k];
    }
    intra /= nfg;
    float inter = 0.0f;
    for (int k1 = 1; k1 < NCLS; ++k1)
      for (int k2 = 1; k2 < NCLS; ++k2)
        inter += (Glds[k1][k1] + Glds[k2][k2] - 2.0f * Glds[k1][k2]) * mask[k1] * mask[k2];
    inter /= (32.0f * nfg * nfg);
    out[n] = intra - inter;
  }
}

extern "C" void kernel_launch(void* const* d_in, const int* in_sizes, int n_in,
                              void* d_out, int out_size, void* d_ws, size_t ws_size,
                              hipStream_t stream) {
  const float* emb = (const float*)d_in[0];
  const int*   lab = (const int*)d_in[1];
  float* out = (float*)d_out;
  float* ws  = (float*)d_ws;
  zero_ws_kernel<<<(WS_TOTAL + 255) / 256, 256, 0, stream>>>(ws);
  accum_kernel<<<NIMG * (OUT_H * OUT_W) / 256, 256, 0, stream>>>(emb, lab, ws);
  finalize_kernel<<<NIMG, 32, 0, stream>>>(ws, out);
}